// MoEAutoencoder_44049184588242
// MI455X (gfx1250) — compile-verified
//
#include <hip/hip_runtime.h>
#include <hip/hip_bf16.h>

typedef float v2f __attribute__((ext_vector_type(2)));
typedef float v8f __attribute__((ext_vector_type(8)));

#define N_ROWS 8192
#define D_IN   2048
#define H_DIM  1024
#define E_NUM  8
#define SEG_AL 128                      // expert segment alignment = BM tile
#define NGMAX  (N_ROWS + E_NUM * SEG_AL)   // 9216

// ---------------------------------------------------------------------------
// Tiled FP32 WMMA GEMM: C[M,N] = A[M,K] @ B[K,N] (+bias) (+relu)
// block = 128 threads (4 waves); block tile 128x64, BK=16.
// Each wave owns a 32x64 strip -> 8 independent v8f accumulator chains.
// LDS holds K-pair-packed fragments (v2f) so every WMMA fragment is one
// ds_load_b64 into an adjacent VGPR pair. Register double-buffering overlaps
// next tile's global_load_b128 with the current WMMA burst.
// EXPERT mode: B = expert_W[e]; e resolved per 128-row block from segbase[].
// ---------------------------------------------------------------------------
template<bool RELU, bool BIAS, bool EXPERT>
__global__ __launch_bounds__(128)
void gemm_f32_wmma(const float* __restrict__ A, const float* __restrict__ B,
                   const float* __restrict__ bias, float* __restrict__ C,
                   int M, int Nn, int K, int lda, int ldb, int ldc,
                   const int* __restrict__ segbase)
{
    __shared__ v2f AsP[8][128];   // AsP[j][m] = { A[m][2j], A[m][2j+1] }   (8 KB)
    __shared__ v2f BsP[8][64];    // BsP[j][n] = { B[2j][n], B[2j+1][n] }   (4 KB)

    const int row0 = blockIdx.y * 128;
    const int col0 = blockIdx.x * 64;

    const float* Bp = B;
    if (EXPERT) {
        int e = 0;
        #pragma unroll
        for (int q = 1; q < E_NUM; ++q)
            if (row0 >= segbase[q]) e = q;
        Bp = B + (size_t)e * (size_t)K * (size_t)ldb;
    }

    const int t     = threadIdx.x;
    const int lane  = t & 31;
    const int w     = t >> 5;           // wave id 0..3 -> 32-row strip
    const int m     = lane & 15;
    const int khalf = (lane >> 4) << 1; // 0 (lanes 0-15) or 2 (lanes 16-31)

    v8f acc[8] = {};                    // [g*4+nt]: g = M-substrip, nt = N-subtile

    // staging registers (double buffer in registers)
    float4 ar[4];                       // A: one row (=t), 16 k values
    float4 br[2];                       // B: row r=t>>3, 8 cols at (t&7)*8

    const int arow = t;                 // 128 rows, one per thread
    const int br_r = t >> 3;            // 0..15
    const int br_c = (t & 7) * 8;       // 0..56

    auto load_global = [&](int kb) {
        const float* ap = A + (size_t)(row0 + arow) * lda + kb;
        ar[0] = *(const float4*)(ap + 0);
        ar[1] = *(const float4*)(ap + 4);
        ar[2] = *(const float4*)(ap + 8);
        ar[3] = *(const float4*)(ap + 12);
        const float* bp = Bp + (size_t)(kb + br_r) * ldb + col0 + br_c;
        br[0] = *(const float4*)(bp + 0);
        br[1] = *(const float4*)(bp + 4);
    };
    auto store_lds = [&]() {
        #pragma unroll
        for (int q = 0; q < 4; ++q) {
            v2f lo; lo.x = ar[q].x; lo.y = ar[q].y;
            v2f hi; hi.x = ar[q].z; hi.y = ar[q].w;
            AsP[2 * q + 0][arow] = lo;
            AsP[2 * q + 1][arow] = hi;
        }
        float* bs = (float*)&BsP[0][0];
        const int jrow = br_r >> 1, par = br_r & 1;
        float v[8] = { br[0].x, br[0].y, br[0].z, br[0].w,
                       br[1].x, br[1].y, br[1].z, br[1].w };
        #pragma unroll
        for (int q = 0; q < 8; ++q)
            bs[((size_t)jrow * 64 + br_c + q) * 2 + par] = v[q];
    };

    load_global(0);
    for (int kb = 0; kb < K; kb += 16) {
        store_lds();
        __syncthreads();
        if (kb + 16 < K) load_global(kb + 16);   // in flight during WMMA burst

        #pragma unroll
        for (int kk = 0; kk < 16; kk += 4) {
            const int j = (kk + khalf) >> 1;     // even k2 -> pair index
            v2f a0 = AsP[j][w * 32 + m];         // M-substrip 0
            v2f a1 = AsP[j][w * 32 + 16 + m];    // M-substrip 1
            #pragma unroll
            for (int nt = 0; nt < 4; ++nt) {
                v2f b = BsP[j][nt * 16 + m];
                acc[nt] = __builtin_amdgcn_wmma_f32_16x16x4_f32(
                    false, a0, false, b, (short)0, acc[nt], false, false);
                acc[4 + nt] = __builtin_amdgcn_wmma_f32_16x16x4_f32(
                    false, a1, false, b, (short)0, acc[4 + nt], false, false);
            }
        }
        __syncthreads();
    }

    // epilogue: VGPR i of C/D holds M = i (lanes 0-15) or i+8 (lanes 16-31)
    const int hi16 = lane >> 4;
    #pragma unroll
    for (int nt = 0; nt < 4; ++nt) {
        const int col = col0 + nt * 16 + m;
        float bv = BIAS ? bias[col] : 0.0f;
        #pragma unroll
        for (int g = 0; g < 2; ++g) {
            #pragma unroll
            for (int i = 0; i < 8; ++i) {
                const int row = row0 + w * 32 + g * 16 + hi16 * 8 + i;
                float v = acc[g * 4 + nt][i] + bv;
                if (RELU) v = fmaxf(v, 0.0f);
                C[(size_t)row * ldc + col] = v;
            }
        }
    }
}

// ---------------------------------------------------------------------------
// Gate + coef: per-row 8 gate logits + 2 coef logits, softmax, first-argmax.
// fp32 exactness keeps the top-1 routing identical to the fp32 reference.
// ---------------------------------------------------------------------------
__global__ __launch_bounds__(256)
void gate_kernel(const float* __restrict__ h, const float* __restrict__ gate_W,
                 const float* __restrict__ coef_W, const float* __restrict__ coef_b,
                 int* __restrict__ idx, float* __restrict__ s0, float* __restrict__ s1)
{
    const int n = blockIdx.x;
    const int t = threadIdx.x;
    const float* hr = h + (size_t)n * H_DIM;

    float acc[10];
    #pragma unroll
    for (int j = 0; j < 10; ++j) acc[j] = 0.0f;

    for (int k = t; k < H_DIM; k += 256) {
        const float hv = hr[k];
        const float* gw = gate_W + (size_t)k * E_NUM;
        #pragma unroll
        for (int e = 0; e < E_NUM; ++e) acc[e] += hv * gw[e];
        acc[8] += hv * coef_W[(size_t)k * 2 + 0];
        acc[9] += hv * coef_W[(size_t)k * 2 + 1];
    }

    __shared__ float red[10][256];
    #pragma unroll
    for (int j = 0; j < 10; ++j) red[j][t] = acc[j];
    __syncthreads();
    for (int s = 128; s > 0; s >>= 1) {
        if (t < s) {
            #pragma unroll
            for (int j = 0; j < 10; ++j) red[j][t] += red[j][t + s];
        }
        __syncthreads();
    }

    if (t == 0) {
        float l[E_NUM];
        #pragma unroll
        for (int e = 0; e < E_NUM; ++e) l[e] = red[e][0];
        int am = 0; float lm = l[0];
        #pragma unroll
        for (int e = 1; e < E_NUM; ++e)
            if (l[e] > lm) { lm = l[e]; am = e; }       // first-max semantics
        float ss = 0.0f;
        #pragma unroll
        for (int e = 0; e < E_NUM; ++e) ss += expf(l[e] - lm);
        const float p = 1.0f / ss;                      // softmax value at argmax

        const float c0l = red[8][0] + coef_b[0];
        const float c1l = red[9][0] + coef_b[1];
        const float cm  = fmaxf(c0l, c1l);
        const float e0  = expf(c0l - cm), e1 = expf(c1l - cm);
        const float inv = 1.0f / (e0 + e1);

        idx[n] = am;
        s0[n]  = p * e0 * inv;   // p * coef0
        s1[n]  = e1 * inv;       // coef1
    }
}

// ---------------------------------------------------------------------------
// MoE dispatch helpers (atomics only permute rows within an expert segment;
// the scattered final result is order-invariant -> deterministic output).
// ---------------------------------------------------------------------------
__global__ void init_kernel(int* __restrict__ perm, int* __restrict__ cnt,
                            int* __restrict__ cursor)
{
    const int i = blockIdx.x * 256 + threadIdx.x;
    if (i < NGMAX) perm[i] = -1;
    if (i < E_NUM) { cnt[i] = 0; cursor[i] = 0; }
}

__global__ void hist_kernel(const int* __restrict__ idx, int* __restrict__ cnt)
{
    const int n = blockIdx.x * 256 + threadIdx.x;
    if (n < N_ROWS) atomicAdd(&cnt[idx[n]], 1);
}

__global__ void scan_kernel(const int* __restrict__ cnt, int* __restrict__ base)
{
    if (threadIdx.x == 0 && blockIdx.x == 0) {
        base[0] = 0;
        for (int e = 0; e < E_NUM; ++e)
            base[e + 1] = base[e] + ((cnt[e] + (SEG_AL - 1)) & ~(SEG_AL - 1));
    }
}

__global__ void perm_kernel(const int* __restrict__ idx, const int* __restrict__ base,
                            int* __restrict__ cursor, int* __restrict__ perm)
{
    const int n = blockIdx.x * 256 + threadIdx.x;
    if (n < N_ROWS) {
        const int e = idx[n];
        const int pos = base[e] + atomicAdd(&cursor[e], 1);
        perm[pos] = n;
    }
}

__global__ __launch_bounds__(256)
void gather_kernel(const float* __restrict__ h, const int* __restrict__ perm,
                   float* __restrict__ hg)
{
    const int pos = blockIdx.x;
    const int n = perm[pos];
    float4* dst = (float4*)(hg + (size_t)pos * H_DIM) + threadIdx.x;
    if (n >= 0) {
        const float4* src = (const float4*)(h + (size_t)n * H_DIM) + threadIdx.x;
        *dst = *src;
    } else {
        float4 z; z.x = z.y = z.z = z.w = 0.0f;
        *dst = z;   // pad rows -> zeros; their GEMM results are never read
    }
}

__global__ __launch_bounds__(256)
void combine_kernel(const float* __restrict__ yg, const float* __restrict__ mlp,
                    const float* __restrict__ expert_b, const int* __restrict__ perm,
                    const int* __restrict__ base, const float* __restrict__ s0,
                    const float* __restrict__ s1, float* __restrict__ mixed)
{
    const int pos = blockIdx.x;
    const int n = perm[pos];
    if (n < 0) return;
    int e = 0;
    #pragma unroll
    for (int q = 1; q < E_NUM; ++q)
        if (pos >= base[q]) e = q;

    const float a = s0[n], b = s1[n];
    const int j = threadIdx.x * 4;
    const float* yp = yg + (size_t)pos * H_DIM + j;
    const float* bp = expert_b + (size_t)e * H_DIM + j;
    const float* mp = mlp + (size_t)n * H_DIM + j;
    float* op = mixed + (size_t)n * H_DIM + j;
    #pragma unroll
    for (int q = 0; q < 4; ++q)
        op[q] = a * (yp[q] + bp[q]) + b * mp[q];
}

// ---------------------------------------------------------------------------
extern "C" void kernel_launch(void* const* d_in, const int* in_sizes, int n_in,
                              void* d_out, int out_size, void* d_ws, size_t ws_size,
                              hipStream_t stream)
{
    const float* x        = (const float*)d_in[0];
    const float* enc_W    = (const float*)d_in[1];
    const float* enc_b    = (const float*)d_in[2];
    const float* gate_W   = (const float*)d_in[3];
    const float* expert_W = (const float*)d_in[4];
    const float* expert_b = (const float*)d_in[5];
    const float* mlp_W    = (const float*)d_in[6];
    const float* mlp_b    = (const float*)d_in[7];
    const float* coef_W   = (const float*)d_in[8];
    const float* coef_b   = (const float*)d_in[9];
    const float* dec_W    = (const float*)d_in[10];
    const float* dec_b    = (const float*)d_in[11];
    float* out = (float*)d_out;

    // workspace layout (256B aligned slabs)
    char* ws = (char*)d_ws;
    size_t off = 0;
    auto take = [&](size_t bytes) -> void* {
        void* p = ws + off;
        off = (off + bytes + 255) & ~(size_t)255;
        return p;
    };
    float* h      = (float*)take((size_t)N_ROWS * H_DIM * 4);   // 32 MB
    float* hg     = (float*)take((size_t)NGMAX  * H_DIM * 4);   // 36 MB (reused as `mixed`)
    float* yg     = (float*)take((size_t)NGMAX  * H_DIM * 4);   // 36 MB
    float* mlpo   = (float*)take((size_t)N_ROWS * H_DIM * 4);   // 32 MB
    int*   idx    = (int*)  take((size_t)N_ROWS * 4);
    float* s0     = (float*)take((size_t)N_ROWS * 4);
    float* s1     = (float*)take((size_t)N_ROWS * 4);
    int*   perm   = (int*)  take((size_t)NGMAX * 4);
    int*   cnt    = (int*)  take(E_NUM * 4);
    int*   base   = (int*)  take((E_NUM + 1) * 4);
    int*   cursor = (int*)  take(E_NUM * 4);
    (void)ws_size; (void)in_sizes; (void)n_in; (void)out_size;

    const dim3 blk128(128), blk256(256);

    // 1) h = relu(x @ enc_W + enc_b)      [8192,2048]@[2048,1024]
    gemm_f32_wmma<true, true, false><<<dim3(H_DIM / 64, N_ROWS / 128), blk128, 0, stream>>>(
        x, enc_W, enc_b, h, N_ROWS, H_DIM, D_IN, D_IN, H_DIM, H_DIM, nullptr);

    // 2) gate + coef + top-1
    gate_kernel<<<dim3(N_ROWS), blk256, 0, stream>>>(h, gate_W, coef_W, coef_b, idx, s0, s1);

    // 3) dispatch: init -> histogram -> 128-aligned segment scan -> compaction
    init_kernel<<<dim3((NGMAX + 255) / 256), blk256, 0, stream>>>(perm, cnt, cursor);
    hist_kernel<<<dim3(N_ROWS / 256), blk256, 0, stream>>>(idx, cnt);
    scan_kernel<<<dim3(1), dim3(1), 0, stream>>>(cnt, base);
    perm_kernel<<<dim3(N_ROWS / 256), blk256, 0, stream>>>(idx, base, cursor, perm);

    // 4) gather h rows into expert-contiguous hg
    gather_kernel<<<dim3(NGMAX), blk256, 0, stream>>>(h, perm, hg);

    // 5a) mlp_out = h @ mlp_W + mlp_b     [8192,1024]@[1024,1024]
    gemm_f32_wmma<false, true, false><<<dim3(H_DIM / 64, N_ROWS / 128), blk128, 0, stream>>>(
        h, mlp_W, mlp_b, mlpo, N_ROWS, H_DIM, H_DIM, H_DIM, H_DIM, H_DIM, nullptr);

    // 5b) yg = hg @ expert_W[seg(row)]    [9216,1024]@[1024,1024] per segment
    gemm_f32_wmma<false, false, true><<<dim3(H_DIM / 64, NGMAX / 128), blk128, 0, stream>>>(
        hg, expert_W, nullptr, yg, NGMAX, H_DIM, H_DIM, H_DIM, H_DIM, H_DIM, base);

    // 6) mixed[n] = s0*(yg+expert_b) + s1*mlp   (scatter; reuse hg as `mixed`)
    combine_kernel<<<dim3(NGMAX), blk256, 0, stream>>>(yg, mlpo, expert_b, perm, base, s0, s1, hg);

    // 7) out = mixed @ dec_W + dec_b      [8192,1024]@[1024,2048]
    gemm_f32_wmma<false, true, false><<<dim3(D_IN / 64, N_ROWS / 128), blk128, 0, stream>>>(
        hg, dec_W, dec_b, out, N_ROWS, D_IN, H_DIM, H_DIM, D_IN, D_IN, nullptr);
}